// SetOfOrthonormalTransforms_87729001988573
// MI455X (gfx1250) — compile-verified
//
#include <hip/hip_runtime.h>

typedef __attribute__((ext_vector_type(2))) float v2f;
typedef __attribute__((ext_vector_type(8))) float v8f;

#define NMAT   8
#define NBLKS  8192
#define COLS   512
#define NANGS  28
#define BLKSTRIDE (NMAT * COLS)   // 4096 floats per block

// One workgroup: 8 global blocks (4 pairs), 8 waves.
// Phase A: threads 0..63 build the 8 R matrices column-parallel into LDS.
// Phase B: wave w -> pair (w>>1), column half (w&1); 16 tiles of 16 cols each,
//          4 chained V_WMMA_F32_16X16X4_F32 per tile (K=16 over block-diag A).
__global__ __launch_bounds__(256) void ortho_wmma_kernel(
    const float* __restrict__ X, const float* __restrict__ angles,
    const float* __restrict__ mus, float* __restrict__ Y)
{
    __shared__ float Rs[8][8][8];   // [local block][row][col]
    const int tid = threadIdx.x;

    // ---------- Phase A: compose Givens chain, 8 threads per R (one per column)
    if (tid < 64) {
        const int rl  = tid >> 3;          // local block 0..7
        const int col = tid & 7;           // column of R this thread owns
        const int blk = blockIdx.x * 8 + rl;
        const float* __restrict__ ang = angles + blk * NANGS;

        float m[8];
        #pragma unroll
        for (int i = 0; i < 8; ++i) m[i] = (i == col) ? 1.0f : 0.0f;

        int p = 0;
        #pragma unroll
        for (int i = 0; i < 8; ++i) {
            #pragma unroll
            for (int j = i + 1; j < 8; ++j) {
                float s, c;
                sincosf(ang[p], &s, &c);
                ++p;
                const float vt = m[i], vb = m[j];
                m[i] = c * vt - s * vb;
                m[j] = s * vt + c * vb;
            }
        }
        const float* __restrict__ mu = mus + blk * 8;
        #pragma unroll
        for (int i = 0; i < 8; ++i) Rs[rl][i][col] = mu[i] * m[i];
    }
    __syncthreads();

    // ---------- Phase B: WMMA matmul, one wave per (pair, column-half)
    const int wv    = tid >> 5;        // wave 0..7
    const int lane  = tid & 31;
    const int half  = lane >> 4;       // lane half selects K sub-slice / output block
    const int n     = lane & 15;       // B/C column within tile, also A row M%16
    const int prl   = wv >> 1;         // local pair 0..3
    const int chalf = wv & 1;          // which 256-column half
    const int r0    = prl * 2;         // local block index of first block in pair
    const int b0    = blockIdx.x * 8 + r0;   // global block index

    // A: block-diagonal [16x16] of (R_b0, R_b1), sliced into 4 K-chunks of 4.
    // VGPR j of slice s holds A[m][k], k = 4s + j + 2*half, m = lane&15.
    v2f A[4];
    #pragma unroll
    for (int s = 0; s < 4; ++s) {
        #pragma unroll
        for (int j = 0; j < 2; ++j) {
            const int k = 4 * s + j + 2 * half;
            float v = 0.0f;
            if (n < 8) { if (k < 8)  v = Rs[r0][n][k]; }
            else       { if (k >= 8) v = Rs[r0 + 1][n - 8][k - 8]; }
            A[s][j] = v;
        }
    }

    // Per-lane X offsets for the 8 B elements (row k of the stacked 16-row B).
    unsigned xoff[4][2];
    #pragma unroll
    for (int s = 0; s < 4; ++s) {
        #pragma unroll
        for (int j = 0; j < 2; ++j) {
            const int k = 4 * s + j + 2 * half;
            xoff[s][j] = (unsigned)(b0 + (k >> 3)) * BLKSTRIDE
                       + (unsigned)(k & 7) * COLS + (unsigned)n;
        }
    }
    const unsigned ybase = (unsigned)(b0 + half) * BLKSTRIDE + (unsigned)n;
    const int c0 = chalf * 256;

    #pragma unroll 2
    for (int t = 0; t < 16; ++t) {
        const int c = c0 + t * 16;

        v2f B[4];
        #pragma unroll
        for (int s = 0; s < 4; ++s) {
            B[s][0] = __builtin_nontemporal_load(X + xoff[s][0] + c);
            B[s][1] = __builtin_nontemporal_load(X + xoff[s][1] + c);
        }

        v8f acc = {0.f, 0.f, 0.f, 0.f, 0.f, 0.f, 0.f, 0.f};
        acc = __builtin_amdgcn_wmma_f32_16x16x4_f32(false, A[0], false, B[0],
                                                    (short)0, acc, false, false);
        acc = __builtin_amdgcn_wmma_f32_16x16x4_f32(false, A[1], false, B[1],
                                                    (short)0, acc, false, false);
        acc = __builtin_amdgcn_wmma_f32_16x16x4_f32(false, A[2], false, B[2],
                                                    (short)0, acc, false, false);
        acc = __builtin_amdgcn_wmma_f32_16x16x4_f32(false, A[3], false, B[3],
                                                    (short)0, acc, false, false);

        // C/D layout: VGPR r = (row r lanes 0-15 -> Y_b0, row r+8 lanes 16-31 -> Y_b1)
        #pragma unroll
        for (int r = 0; r < 8; ++r)
            __builtin_nontemporal_store(acc[r], Y + ybase + (unsigned)r * COLS + c);
    }
}

extern "C" void kernel_launch(void* const* d_in, const int* in_sizes, int n_in,
                              void* d_out, int out_size, void* d_ws, size_t ws_size,
                              hipStream_t stream) {
    const float* X      = (const float*)d_in[0];  // [8192][8][512] f32
    const float* angles = (const float*)d_in[1];  // [8192][28]     f32
    const float* mus    = (const float*)d_in[2];  // [8192][8]      f32
    float*       Y      = (float*)d_out;          // [8192][8][512] f32
    (void)in_sizes; (void)n_in; (void)out_size; (void)d_ws; (void)ws_size;

    ortho_wmma_kernel<<<NBLKS / 8, 256, 0, stream>>>(X, angles, mus, Y);
}